// Generator_24910810317454
// MI455X (gfx1250) — compile-verified
//
#include <hip/hip_runtime.h>
#include <hip/hip_bf16.h>

typedef float v2f __attribute__((ext_vector_type(2)));
typedef float v8f __attribute__((ext_vector_type(8)));

#define BB 8
#define HH 1024
#define EE 512
#define KV (HH + EE)        // 1536 : x_v length
#define XK (2 * HH + EE)    // 2560 : x_k length
#define LDSS 1540           // LDS row stride in dwords (1540 % 64 == 4 -> conflict-free)

// ---------------------------------------------------------------------------
// Zero-fill workspace (acc floats + cnt u32)
// ---------------------------------------------------------------------------
__global__ void zero_u32(unsigned* __restrict__ p, int n) {
    int i = blockIdx.x * blockDim.x + threadIdx.x;
    if (i < n) p[i] = 0u;
}

// ---------------------------------------------------------------------------
// Dense e_v = x_v @ W_V^T + b_V  via V_WMMA_F32_16X16X4_F32.
// One wave computes a 16(batch-padded) x 16(words) tile over K=1536.
// A (16x4 f32): lane L holds M=L%16, K=(L/16)*2+j  -> from LDS (zero row for M>=8)
// B (4x16 f32): lane L holds N=L%16, K=(L/16)*2+j  -> streamed from W_V
// D (16x16 f32): vgpr r, lanes 0-15: M=r, N=lane   -> rows r<8 are real batches
// ---------------------------------------------------------------------------
__global__ __launch_bounds__(256) void ev_gemm(const float* __restrict__ output,
                                               const float* __restrict__ input_step,
                                               const float* __restrict__ WV,
                                               const float* __restrict__ bV,
                                               float* __restrict__ ev, int V) {
    __shared__ float xs[9 * LDSS];

    // Stage x_v = concat(output, input_step) rows 0..7; row 8 = zeros (padding source).
    for (int i = threadIdx.x; i < BB * KV; i += 256) {
        int m = i / KV, k = i - m * KV;
        float val = (k < HH) ? output[m * HH + k] : input_step[m * EE + (k - HH)];
        xs[m * LDSS + k] = val;
    }
    for (int k = threadIdx.x; k < KV; k += 256) xs[8 * LDSS + k] = 0.0f;
    __syncthreads();

    const int lane  = threadIdx.x & 31;
    const int wave  = threadIdx.x >> 5;
    const int tile  = blockIdx.x * 8 + wave;
    const int wbase = tile * 16;
    if (wbase + 16 > V) return;                 // uniform per-wave guard

    const int m    = lane & 15;
    const int meff = (m < 8) ? m : 8;           // padding lanes read the zero row
    const int koff = (lane >> 4) * 2;           // 0 or 2

    const float* __restrict__ arow = &xs[meff * LDSS + koff];
    const float* __restrict__ brow = WV + ((size_t)(wbase + m) * KV + koff);

    v8f c = {0.f, 0.f, 0.f, 0.f, 0.f, 0.f, 0.f, 0.f};
#pragma unroll 8
    for (int kb = 0; kb < KV; kb += 4) {
        v2f a = *(const v2f*)(arow + kb);       // ds_load_b64
        v2f b = *(const v2f*)(brow + kb);       // global_load_b64 (streams W_V once)
        c = __builtin_amdgcn_wmma_f32_16x16x4_f32(false, a, false, b,
                                                  (short)0, c, false, false);
    }

    const float bias = bV[wbase + m];
    if (lane < 16) {
#pragma unroll
        for (int r = 0; r < BB; ++r)
            ev[(size_t)r * V + wbase + lane] = c[r] + bias;   // 16 contiguous floats
    }
}

// ---------------------------------------------------------------------------
// Sparse e_k: one block per (b,t) pair. Computes dot(x_k[b], W_K[idx]) + b_K[idx]
// and atomically accumulates into acc[b,idx]; per-occurrence add => k * e_k.
// ---------------------------------------------------------------------------
__global__ __launch_bounds__(256) void ek_sparse(const float* __restrict__ output,
                                                 const float* __restrict__ input_step,
                                                 const float* __restrict__ context,
                                                 const int* __restrict__ topic,
                                                 const float* __restrict__ WK,
                                                 const float* __restrict__ bK,
                                                 float* __restrict__ acc,
                                                 unsigned* __restrict__ cnt,
                                                 int V, int T) {
    const int b = blockIdx.x / T;
    const int t = blockIdx.x - b * T;
    const int idx = topic[b * T + t];
    if (idx == 0) return;                       // pad word contributes nothing

    __shared__ float red[256];
    const float* __restrict__ wrow = WK + (size_t)idx * XK;

    float partial = 0.0f;
    for (int c = threadIdx.x; c < XK; c += 256) {
        float xv;
        if (c < HH)            xv = output[b * HH + c];
        else if (c < HH + EE)  xv = input_step[b * EE + (c - HH)];
        else                   xv = context[b * HH + (c - HH - EE)];
        partial = fmaf(xv, wrow[c], partial);
    }
    red[threadIdx.x] = partial;
    __syncthreads();
    for (int s = 128; s > 0; s >>= 1) {
        if (threadIdx.x < s) red[threadIdx.x] += red[threadIdx.x + s];
        __syncthreads();
    }
    if (threadIdx.x == 0) {
        atomicAdd(&acc[(size_t)b * V + idx], red[0] + bK[idx]);
        atomicAdd(&cnt[(size_t)b * V + idx], 1u);
    }
}

// ---------------------------------------------------------------------------
// energy = tanh(cnt ? acc : e_v), then stable softmax over V. One block per row.
// ---------------------------------------------------------------------------
__global__ __launch_bounds__(256) void combine_softmax(float* __restrict__ eout,
                                                       const float* __restrict__ acc,
                                                       const unsigned* __restrict__ cnt,
                                                       int V) {
    const int b = blockIdx.x;
    const int tid = threadIdx.x;
    const size_t base = (size_t)b * V;
    __shared__ float red[256];

    float lmax = -3.0e38f;
    for (int w = tid; w < V; w += 256) {
        float evv = eout[base + w];
        float raw = cnt[base + w] ? acc[base + w] : evv;
        float e = tanhf(raw);
        eout[base + w] = e;
        lmax = fmaxf(lmax, e);
    }
    red[tid] = lmax;
    __syncthreads();
    for (int s = 128; s > 0; s >>= 1) {
        if (tid < s) red[tid] = fmaxf(red[tid], red[tid + s]);
        __syncthreads();
    }
    const float gmax = red[0];
    __syncthreads();

    float lsum = 0.0f;
    for (int w = tid; w < V; w += 256) {
        float e = __expf(eout[base + w] - gmax);
        eout[base + w] = e;
        lsum += e;
    }
    red[tid] = lsum;
    __syncthreads();
    for (int s = 128; s > 0; s >>= 1) {
        if (tid < s) red[tid] += red[tid + s];
        __syncthreads();
    }
    const float inv = 1.0f / red[0];
    __syncthreads();

    for (int w = tid; w < V; w += 256)
        eout[base + w] *= inv;
}

// ---------------------------------------------------------------------------
extern "C" void kernel_launch(void* const* d_in, const int* in_sizes, int n_in,
                              void* d_out, int out_size, void* d_ws, size_t ws_size,
                              hipStream_t stream) {
    const float* output     = (const float*)d_in[0];   // [1,8,1024]
    const float* input_step = (const float*)d_in[1];   // [1,8,512]
    const float* context    = (const float*)d_in[2];   // [1,8,1024]
    const int*   topic      = (const int*)d_in[3];     // [8,T]
    const float* WV         = (const float*)d_in[5];   // [V,1536]
    const float* bV         = (const float*)d_in[6];   // [V]
    const float* WK         = (const float*)d_in[7];   // [V,2560]
    const float* bK         = (const float*)d_in[8];   // [V]

    const int V = in_sizes[6];          // 32000
    const int T = in_sizes[3] / BB;     // 100

    float*    acc = (float*)d_ws;                            // [8,V] f32
    unsigned* cnt = (unsigned*)(acc + (size_t)BB * V);       // [8,V] u32
    float*    out = (float*)d_out;                           // [8,V]

    // 1) zero acc + cnt
    const int nz = 2 * BB * V;
    zero_u32<<<(nz + 255) / 256, 256, 0, stream>>>((unsigned*)d_ws, nz);

    // 2) dense e_v GEMM (WMMA f32), writes e_v + bias into d_out
    const int tiles  = (V + 15) / 16;          // 2000
    const int blocks = (tiles + 7) / 8;        // 250 (8 waves/block)
    ev_gemm<<<blocks, 256, 0, stream>>>(output, input_step, WV, bV, out, V);

    // 3) sparse k * e_k accumulation (<= 800 gathered W_K rows)
    ek_sparse<<<BB * T, 256, 0, stream>>>(output, input_step, context, topic,
                                          WK, bK, acc, cnt, V, T);

    // 4) tanh combine + softmax per batch row
    combine_softmax<<<BB, 256, 0, stream>>>(out, acc, cnt, V);
}